// GNNPolicy_55482387529962
// MI455X (gfx1250) — compile-verified
//
#include <hip/hip_runtime.h>
#include <hip/hip_bf16.h>

#define NN 4096
#define HDIM 128
#define KH 4
#define HDH 32

typedef __attribute__((ext_vector_type(2))) float v2f;
typedef __attribute__((ext_vector_type(8))) float v8f;

// ---------------- K0: x = relu(nf @ Wi + bi) ----------------
__global__ void k_input_proj(const float* __restrict__ nf, const float* __restrict__ Wi,
                             const float* __restrict__ bi, float* __restrict__ x) {
    int n = blockIdx.x, t = threadIdx.x;                 // 128 threads
    float acc = bi[t];
#pragma unroll
    for (int d = 0; d < 5; ++d) acc = fmaf(nf[n * 5 + d], Wi[d * HDIM + t], acc);
    x[(size_t)n * HDIM + t] = fmaxf(acc, 0.f);
}

// ---------------- K2: edge bitmask (adj>0 || i==j), 1 bit per edge ----------------
__global__ void k_mask_words(const float* __restrict__ adj, unsigned* __restrict__ mw) {
    int g = blockIdx.x * blockDim.x + threadIdx.x;       // NN*128 words total
    if (g >= NN * (NN / 32)) return;
    int i = g >> 7, w = g & 127;
    const float* row = adj + (size_t)i * NN + w * 32;
    unsigned bits = 0u;
#pragma unroll
    for (int b = 0; b < 32; ++b) {
        int j = w * 32 + b;
        if (row[b] > 0.f || j == i) bits |= (1u << b);
    }
    mw[g] = bits;
}

// ---------------- K1: Wx = x @ W2 via V_WMMA_F32_16X16X4_F32 ----------------
// gatW points at gat_W[l]: (K, H, HD).  W2[d][k*32+h] = gatW[k][d][h].
// One wave computes a 16x16 tile of the (4096 x 128) output.
__global__ void k_wx_gemm(const float* __restrict__ x, const float* __restrict__ gatW,
                          float* __restrict__ Wx) {
    int i0 = blockIdx.x * 16, h0 = blockIdx.y * 16;
    int lane = threadIdx.x;
    int r = lane & 15;                    // A: row M, B: column N
    int sld = (lane < 16) ? 0 : 2;        // K-slot base per half-wave
    int hg = h0 + r;
    int kh = hg >> 5, hl = hg & 31;
    const float* xrow = x + (size_t)(i0 + r) * HDIM;
    const float* wcol = gatW + (size_t)kh * HDIM * HDH + hl;   // stride HDH over d
    v8f c = {};
    for (int kc = 0; kc < HDIM; kc += 4) {
        v2f a, b;
        a.x = xrow[kc + sld];
        a.y = xrow[kc + sld + 1];
        b.x = wcol[(size_t)(kc + sld) * HDH];
        b.y = wcol[(size_t)(kc + sld + 1) * HDH];
        c = __builtin_amdgcn_wmma_f32_16x16x4_f32(false, a, false, b, (short)0, c, false, false);
    }
    // C layout: VGPR v -> row i0+v (lanes<16) / i0+8+v (lanes>=16), col = lane&15
    int hgc = h0 + r;
    int kk = hgc >> 5, hlc = hgc & 31;
    int ibase = i0 + ((lane < 16) ? 0 : 8);
#pragma unroll
    for (int v = 0; v < 8; ++v)
        Wx[((size_t)kk * NN + (ibase + v)) * HDH + hlc] = c[v];
}

// ---------------- K1b: a_src/a_dst = Wx . gat_a halves ----------------
__global__ void k_attn_coef(const float* __restrict__ Wx, const float* __restrict__ gat_a,
                            float* __restrict__ a_src, float* __restrict__ a_dst) {
    int n = blockIdx.x, k = blockIdx.y, lane = threadIdx.x;  // 32 threads
    float w = Wx[((size_t)k * NN + n) * HDH + lane];
    float ps = w * gat_a[k * 64 + lane];
    float pd = w * gat_a[k * 64 + 32 + lane];
    for (int off = 16; off; off >>= 1) {
        ps += __shfl_xor(ps, off, 32);
        pd += __shfl_xor(pd, off, 32);
    }
    if (lane == 0) { a_src[k * NN + n] = ps; a_dst[k * NN + n] = pd; }
}

// ---------------- K3: per-row softmax stats (max via monotone lrelu, then Z) ----------------
__global__ void k_softmax_stats(const float* __restrict__ a_src, const float* __restrict__ a_dst,
                                const unsigned* __restrict__ mw,
                                float* __restrict__ mbuf, float* __restrict__ zinv) {
    int i = blockIdx.x, k = blockIdx.y, lane = threadIdx.x;  // 32 threads
    const unsigned* row = mw + (size_t)i * 128;
    const float* d = a_dst + (size_t)k * NN;
    float s = a_src[k * NN + i];
    float dmax = -3.0e38f;
    for (int t = 0; t < 128; ++t) {
        unsigned w = row[t];
        if ((w >> lane) & 1u) dmax = fmaxf(dmax, d[t * 32 + lane]);
    }
    for (int off = 16; off; off >>= 1) dmax = fmaxf(dmax, __shfl_xor(dmax, off, 32));
    float m = s + dmax; m = (m > 0.f) ? m : 0.2f * m;     // lrelu is monotone -> exact row max
    float z = 0.f;
    for (int t = 0; t < 128; ++t) {
        unsigned w = row[t];
        if ((w >> lane) & 1u) {
            float e = s + d[t * 32 + lane]; e = (e > 0.f) ? e : 0.2f * e;
            z += __expf(e - m);
        }
    }
    for (int off = 16; off; off >>= 1) z += __shfl_xor(z, off, 32);
    if (lane == 0) { mbuf[k * NN + i] = m; zinv[k * NN + i] = 1.f / z; }
}

// ---------------- K4: h = elu(alpha @ Wx) via V_WMMA_F32_16X16X4_F32 ----------------
// 4-wave workgroup per (head k, 16-row tile). Each wave owns a 1024-wide j slice;
// a_dst row (16KB) and the 16 mask-word rows (8KB) are staged in LDS; partial C
// tiles are reduced through LDS at the end (8KB). Total LDS = 32KB.
__global__ void __launch_bounds__(128) k_gat_agg(
        const float* __restrict__ Wx, const float* __restrict__ a_src,
        const float* __restrict__ a_dst, const float* __restrict__ mbuf,
        const float* __restrict__ zinv, const unsigned* __restrict__ mw,
        float* __restrict__ hbuf) {
    __shared__ float d_lds[NN];                 // 16 KB
    __shared__ unsigned mw_lds[16 * 128];       // 8 KB
    __shared__ float c_red[4][16 * 32];         // 8 KB
    int i0 = blockIdx.x * 16, k = blockIdx.y;
    int tid = threadIdx.x;
    int wave = tid >> 5, lane = tid & 31;

    // cooperative staging (coalesced)
    for (int j = tid; j < NN; j += 128) d_lds[j] = a_dst[(size_t)k * NN + j];
    for (int t = tid; t < 16 * 128; t += 128)
        mw_lds[t] = mw[(size_t)(i0 + (t >> 7)) * 128 + (t & 127)];
    __syncthreads();

    int r = lane & 15;
    int sld = (lane < 16) ? 0 : 2;
    int i = i0 + r;
    float s  = a_src[k * NN + i];
    float m  = mbuf[k * NN + i];
    float zi = zinv[k * NN + i];
    const float* wxk = Wx + (size_t)k * NN * HDH;
    v8f c0 = {}, c1 = {};

    for (int t = wave * 32; t < wave * 32 + 32; ++t) {   // 1024 j's per wave
        unsigned w = mw_lds[r * 128 + t];
        int jb = t * 32;
#pragma unroll
        for (int c8 = 0; c8 < 8; ++c8) {
            int j0 = jb + c8 * 4;
            int ja = j0 + sld, jc = ja + 1;
            unsigned ba = (w >> (c8 * 4 + sld)) & 1u;
            unsigned bb = (w >> (c8 * 4 + sld + 1)) & 1u;
            float ea = s + d_lds[ja]; ea = (ea > 0.f) ? ea : 0.2f * ea;
            float eb = s + d_lds[jc]; eb = (eb > 0.f) ? eb : 0.2f * eb;
            v2f a;
            a.x = ba ? __expf(ea - m) * zi : 0.f;   // normalized alpha in A layout
            a.y = bb ? __expf(eb - m) * zi : 0.f;
            const float* wra = wxk + (size_t)ja * HDH;
            const float* wrb = wxk + (size_t)jc * HDH;
            v2f b0, b1;
            b0.x = wra[r];       b0.y = wrb[r];        // h columns 0..15
            b1.x = wra[16 + r];  b1.y = wrb[16 + r];   // h columns 16..31
            c0 = __builtin_amdgcn_wmma_f32_16x16x4_f32(false, a, false, b0, (short)0, c0, false, false);
            c1 = __builtin_amdgcn_wmma_f32_16x16x4_f32(false, a, false, b1, (short)0, c1, false, false);
        }
    }

    // partial C tiles -> LDS (C layout: VGPR v -> row v / v+8, col = lane&15)
    int rowbase = (lane < 16) ? 0 : 8;
#pragma unroll
    for (int v = 0; v < 8; ++v) {
        c_red[wave][(rowbase + v) * 32 + r]      = c0[v];
        c_red[wave][(rowbase + v) * 32 + 16 + r] = c1[v];
    }
    __syncthreads();

    // cross-wave reduce, ELU, store concat-head layout hbuf[n][k*32 + h]
    for (int e = tid; e < 16 * 32; e += 128) {
        float vsum = c_red[0][e] + c_red[1][e] + c_red[2][e] + c_red[3][e];
        vsum = (vsum > 0.f) ? vsum : (__expf(vsum) - 1.f);
        int row = e >> 5, col = e & 31;
        hbuf[(size_t)(i0 + row) * HDIM + k * HDH + col] = vsum;
    }
}

// ---------------- K5: x = LayerNorm(hbuf + x) ----------------
__global__ void k_residual_ln(const float* __restrict__ hbuf, const float* __restrict__ g,
                              const float* __restrict__ b, float* __restrict__ x) {
    __shared__ float red[HDIM];
    int n = blockIdx.x, t = threadIdx.x;                 // 128 threads
    float val = hbuf[(size_t)n * HDIM + t] + x[(size_t)n * HDIM + t];
    red[t] = val; __syncthreads();
    for (int off = 64; off; off >>= 1) { if (t < off) red[t] += red[t + off]; __syncthreads(); }
    float mu = red[0] * (1.f / HDIM); __syncthreads();
    float dv = (val - mu) * (val - mu);
    red[t] = dv; __syncthreads();
    for (int off = 64; off; off >>= 1) { if (t < off) red[t] += red[t + off]; __syncthreads(); }
    float var = red[0] * (1.f / HDIM);
    x[(size_t)n * HDIM + t] = (val - mu) * rsqrtf(var + 1e-5f) * g[t] + b[t];
}

// ---------------- K6: policy logits + pooling scores ----------------
__global__ void k_heads(const float* __restrict__ x, const int* __restrict__ mask,
                        const float* __restrict__ pW1, const float* __restrict__ pb1,
                        const float* __restrict__ pW2, const float* __restrict__ pb2,
                        const float* __restrict__ poolW, const float* __restrict__ poolb,
                        const float* __restrict__ poolv,
                        float* __restrict__ logits, float* __restrict__ scores) {
    __shared__ float xs[HDIM];
    __shared__ float red[HDIM];
    int n = blockIdx.x, t = threadIdx.x;                 // 128 threads
    xs[t] = x[(size_t)n * HDIM + t]; __syncthreads();
    float pcon = 0.f;
    if (t < 64) {
        float acc = pb1[t];
        for (int d = 0; d < HDIM; ++d) acc = fmaf(xs[d], pW1[d * 64 + t], acc);
        pcon = fmaxf(acc, 0.f) * pW2[t];
    }
    red[t] = pcon; __syncthreads();
    for (int off = 64; off; off >>= 1) { if (t < off) red[t] += red[t + off]; __syncthreads(); }
    if (t == 0) logits[n] = mask[n] ? (red[0] + pb2[0]) : -__builtin_inff();
    __syncthreads();
    float acc = poolb[t];
    for (int d = 0; d < HDIM; ++d) acc = fmaf(xs[d], poolW[d * HDIM + t], acc);
    red[t] = tanhf(acc) * poolv[t]; __syncthreads();
    for (int off = 64; off; off >>= 1) { if (t < off) red[t] += red[t + off]; __syncthreads(); }
    if (t == 0) scores[n] = mask[n] ? red[0] : -__builtin_inff();
}

// ---------------- K7: attention pool + value head (single block) ----------------
__global__ void k_pool_value(const float* __restrict__ x, const float* __restrict__ scores,
                             const float* __restrict__ vW1, const float* __restrict__ vb1,
                             const float* __restrict__ vW2, const float* __restrict__ vb2,
                             float* __restrict__ out_value) {
    __shared__ float beta[NN];
    __shared__ float red[HDIM];
    __shared__ float pooled[HDIM];
    int t = threadIdx.x;                                  // 128 threads
    float mx = -__builtin_inff();
    for (int n = t; n < NN; n += HDIM) mx = fmaxf(mx, scores[n]);
    red[t] = mx; __syncthreads();
    for (int off = 64; off; off >>= 1) { if (t < off) red[t] = fmaxf(red[t], red[t + off]); __syncthreads(); }
    mx = red[0]; __syncthreads();
    float zs = 0.f;
    for (int n = t; n < NN; n += HDIM) { float e = __expf(scores[n] - mx); beta[n] = e; zs += e; }
    red[t] = zs; __syncthreads();
    for (int off = 64; off; off >>= 1) { if (t < off) red[t] += red[t + off]; __syncthreads(); }
    float zinv = 1.f / red[0]; __syncthreads();
    float acc = 0.f;
    for (int n = 0; n < NN; ++n) acc = fmaf(beta[n] * zinv, x[(size_t)n * HDIM + t], acc);
    pooled[t] = acc; __syncthreads();
    float vcon = 0.f;
    if (t < 64) {
        float a2 = vb1[t];
        for (int d = 0; d < HDIM; ++d) a2 = fmaf(pooled[d], vW1[d * 64 + t], a2);
        vcon = fmaxf(a2, 0.f) * vW2[t];
    }
    red[t] = vcon; __syncthreads();
    for (int off = 64; off; off >>= 1) { if (t < off) red[t] += red[t + off]; __syncthreads(); }
    if (t == 0) out_value[0] = red[0] + vb2[0];
}

extern "C" void kernel_launch(void* const* d_in, const int* in_sizes, int n_in,
                              void* d_out, int out_size, void* d_ws, size_t ws_size,
                              hipStream_t stream) {
    const float* nf    = (const float*)d_in[0];
    const float* adj   = (const float*)d_in[1];
    const int*   mask  = (const int*)d_in[2];
    const float* Wi    = (const float*)d_in[3];
    const float* bi    = (const float*)d_in[4];
    const float* gat_W = (const float*)d_in[5];
    const float* gat_a = (const float*)d_in[6];
    const float* ln_g  = (const float*)d_in[7];
    const float* ln_b  = (const float*)d_in[8];
    const float* pW1   = (const float*)d_in[9];
    const float* pb1   = (const float*)d_in[10];
    const float* pW2   = (const float*)d_in[11];
    const float* pb2   = (const float*)d_in[12];
    const float* poolW = (const float*)d_in[13];
    const float* poolb = (const float*)d_in[14];
    const float* poolv = (const float*)d_in[15];
    const float* vW1   = (const float*)d_in[16];
    const float* vb1   = (const float*)d_in[17];
    const float* vW2   = (const float*)d_in[18];
    const float* vb2   = (const float*)d_in[19];

    char* ws = (char*)d_ws;
    float*    x     = (float*)(ws);                        // 2 MB
    float*    hbuf  = (float*)(ws + (2u << 20));           // 2 MB
    float*    Wx    = (float*)(ws + (4u << 20));           // 2 MB (K,N,HD)
    unsigned* mw    = (unsigned*)(ws + (6u << 20));        // 2 MB bitmask
    float*    a_src = (float*)(ws + (8u << 20));           // 64 KB
    float*    a_dst = a_src + KH * NN;
    float*    mbuf  = a_dst + KH * NN;
    float*    zinv  = mbuf + KH * NN;
    float*    scores = zinv + KH * NN;                     // 16 KB

    float* logits = (float*)d_out;        // 4096
    float* value  = logits + NN;          // 1

    k_input_proj<<<NN, HDIM, 0, stream>>>(nf, Wi, bi, x);
    k_mask_words<<<(NN * (NN / 32)) / 256, 256, 0, stream>>>(adj, mw);

    for (int l = 0; l < 3; ++l) {
        const float* gWl = gat_W + (size_t)l * KH * HDIM * HDH;
        const float* gal = gat_a + (size_t)l * KH * 64;
        k_wx_gemm<<<dim3(NN / 16, HDIM / 16), 32, 0, stream>>>(x, gWl, Wx);
        k_attn_coef<<<dim3(NN, KH), 32, 0, stream>>>(Wx, gal, a_src, a_dst);
        k_softmax_stats<<<dim3(NN, KH), 32, 0, stream>>>(a_src, a_dst, mw, mbuf, zinv);
        k_gat_agg<<<dim3(NN / 16, KH), 128, 0, stream>>>(Wx, a_src, a_dst, mbuf, zinv, mw, hbuf);
        k_residual_ln<<<NN, HDIM, 0, stream>>>(hbuf, ln_g + l * HDIM, ln_b + l * HDIM, x);
    }

    k_heads<<<NN, HDIM, 0, stream>>>(x, mask, pW1, pb1, pW2, pb2, poolW, poolb, poolv,
                                     logits, scores);
    k_pool_value<<<1, HDIM, 0, stream>>>(x, scores, vW1, vb1, vW2, vb2, value);
}